// PromptQueryFusionBlock_56272661512220
// MI455X (gfx1250) — compile-verified
//
#include <hip/hip_runtime.h>

// ---- problem constants ----
#define Cc   256
#define Pp   256
#define NHh  8
#define HDd  32
#define Bb_  2
#define Hh   48
#define Ww   48
#define Nn   (Hh*Ww)       // 2304
#define Ll   77
#define Kk   80
#define Ii   100
#define C4   1024
#define MAPC (Cc+6)        // 262
#define KTOT (MAPC*9)      // 2358
#define PH   50            // padded map height/width
#define PA   (PH*PH)       // 2500
#define ICP  272           // MAPC padded to multiple of 16

typedef __attribute__((ext_vector_type(2))) float v2f;
typedef __attribute__((ext_vector_type(8))) float v8f;

#if defined(__HIP_DEVICE_COMPILE__)
#if !__has_builtin(__builtin_amdgcn_wmma_f32_16x16x4_f32)
#error "missing __builtin_amdgcn_wmma_f32_16x16x4_f32 for gfx1250 (device pass)"
#endif
#endif

// ---------------- LayerNorm over C (=256 threads/row) ----------------
__global__ void ln_kernel(const float* __restrict__ in, const float* __restrict__ g,
                          const float* __restrict__ bta, float* __restrict__ out, int mode)
{
    __shared__ float red[256];
    int r = blockIdx.x;
    int c = threadIdx.x;
    long idx;
    if (mode == 1) { int b = r / Nn, hw = r % Nn; idx = ((long)b * Cc + c) * Nn + hw; }
    else           { idx = (long)r * Cc + c; }
    float v = in[idx];
    red[c] = v; __syncthreads();
    for (int s = 128; s > 0; s >>= 1) { if (c < s) red[c] += red[c + s]; __syncthreads(); }
    float mean = red[0] * (1.0f / Cc);
    __syncthreads();
    float d = v - mean;
    red[c] = d * d; __syncthreads();
    for (int s = 128; s > 0; s >>= 1) { if (c < s) red[c] += red[c + s]; __syncthreads(); }
    float var = red[0] * (1.0f / Cc);
    out[(long)r * Cc + c] = d * rsqrtf(var + 1e-5f) * g[c] + bta[c];
}

// ---------------- weight transpose: in (K,N) row-major -> out (N,K) ----------------
__global__ void transpose_kernel(const float* __restrict__ in, float* __restrict__ out,
                                 int K, int N)
{
    long e = (long)blockIdx.x * blockDim.x + threadIdx.x;
    if (e >= (long)K * N) return;
    int n = (int)(e % N);
    int k = (int)(e / N);
    out[(long)n * K + k] = in[e];
}

// ---------------- fp32 WMMA GEMM, B pre-transposed (N,Kd) ----------------
// One wave -> one 16x16 tile. Out-of-range rows/cols load CLAMPED addresses
// (junk lands only in rows/cols never stored) -> branch-free K loop, all
// fragment loads are contiguous b64. Requires Kd % 16 == 0.
__global__ void gemm_wmma(const float* __restrict__ A, const float* __restrict__ Bt,
                          const float* __restrict__ bias, const float* __restrict__ resid,
                          float* __restrict__ D,
                          int M, int N, int Kd, int act,
                          long sAb, long sBb, long sDb)
{
    int lane = threadIdx.x;          // wave32
    int hl   = lane >> 4;
    int l16  = lane & 15;
    int tN = blockIdx.x, tM = blockIdx.y, bt = blockIdx.z;
    const float* Ab = A  + (long)bt * sAb;
    const float* Bp = Bt + (long)bt * sBb;
    float*       Dp = D  + (long)bt * sDb;
    int row = tM * 16 + l16;
    int col = tN * 16 + l16;
    int rowc = row < M ? row : M - 1;
    int colc = col < N ? col : N - 1;
    const float* Arow = Ab + (long)rowc * Kd;
    const float* Bcol = Bp + (long)colc * Kd;
    v8f acc = {};
    for (int k0 = 0; k0 < Kd; k0 += 16) {
        v2f a[4], b[4];
#pragma unroll
        for (int u = 0; u < 4; ++u) {
            int ka = k0 + u * 4 + hl * 2;
            a[u] = *(const v2f*)(Arow + ka);
            b[u] = *(const v2f*)(Bcol + ka);
        }
#pragma unroll
        for (int u = 0; u < 4; ++u)
            acc = __builtin_amdgcn_wmma_f32_16x16x4_f32(false, a[u], false, b[u], (short)0,
                                                        acc, false, false);
    }
    if (col >= N) return;
    float bv = bias ? bias[col] : 0.0f;
#pragma unroll
    for (int j = 0; j < 8; ++j) {
        int r = tM * 16 + hl * 8 + j;
        if (r < M) {
            float v = acc[j] + bv;
            if (resid) v += resid[(long)bt * sDb + (long)r * N + col];
            if (act) v = fmaxf(v, 0.0f);
            Dp[(long)r * N + col] = v;
        }
    }
}

// ---------------- attention: one block per (b, pixel), one wave per head ----------------
__global__ void attn_kernel(const float* __restrict__ Q, const float* __restrict__ Kt,
                            const float* __restrict__ V, const unsigned char* __restrict__ pmask,
                            float* __restrict__ ao)
{
    __shared__ float sQ[NHh][HDd];
    __shared__ float sS[NHh][Ll + 3];
    int row = blockIdx.x;
    int b = row / Nn;
    int t = threadIdx.x;             // 256 = 8 waves
    int h = t >> 5, l = t & 31;
    sQ[h][l] = Q[(long)row * Cc + h * HDd + l];
    __syncthreads();
    float myS[3];
    float smax = -1e30f;
#pragma unroll
    for (int jj = 0; jj < 3; ++jj) {
        int j = l + jj * 32;
        float s = -1e30f;
        if (j < Ll) {
            float a = 0.0f;
            const float* kr = Kt + ((long)(b * Ll + j)) * Cc + h * HDd;
            for (int d = 0; d < HDd; ++d) a += sQ[h][d] * kr[d];
            s = pmask[b * Ll + j] ? a * 0.17677669529663687f : -1e9f; // 1/sqrt(32)
        }
        myS[jj] = s;
        smax = fmaxf(smax, s);
    }
    for (int m = 16; m > 0; m >>= 1) smax = fmaxf(smax, __shfl_xor(smax, m, 32));
    float ssum = 0.0f;
#pragma unroll
    for (int jj = 0; jj < 3; ++jj) {
        int j = l + jj * 32;
        if (j < Ll) { float p = expf(myS[jj] - smax); sS[h][j] = p; ssum += p; }
    }
    for (int m = 16; m > 0; m >>= 1) ssum += __shfl_xor(ssum, m, 32);
    __syncthreads();
    float inv = 1.0f / ssum;
    float a = 0.0f;
    for (int j = 0; j < Ll; ++j)
        a += sS[h][j] * V[((long)(b * Ll + j)) * Cc + h * HDd + l];
    ao[(long)row * Cc + h * HDd + l] = a * inv;
}

// ---------------- row L2-normalize ----------------
__global__ void norm_kernel(const float* __restrict__ in, float* __restrict__ out, int dim)
{
    __shared__ float red[256];
    int r = blockIdx.x, t = threadIdx.x;
    float ss = 0.0f;
    for (int c = t; c < dim; c += 256) { float v = in[(long)r * dim + c]; ss += v * v; }
    red[t] = ss; __syncthreads();
    for (int s = 128; s > 0; s >>= 1) { if (t < s) red[t] += red[t + s]; __syncthreads(); }
    float inv = 1.0f / fmaxf(sqrtf(red[0]), 1e-6f);
    for (int c = t; c < dim; c += 256) out[(long)r * dim + c] = in[(long)r * dim + c] * inv;
}

// ---------------- per-row dot with one vector ----------------
__global__ void rowdot_kernel(const float* __restrict__ A, const float* __restrict__ v,
                              float* __restrict__ out, int dim)
{
    __shared__ float red[256];
    int r = blockIdx.x, t = threadIdx.x;
    float s = 0.0f;
    for (int c = t; c < dim; c += 256) s += A[(long)r * dim + c] * v[c];
    red[t] = s; __syncthreads();
    for (int st = 128; st > 0; st >>= 1) { if (t < st) red[t] += red[t + st]; __syncthreads(); }
    if (t == 0) out[r] = red[0];
}

// ---------------- presence: dot -> mask ? sigmoid(dot - null) : 0 (in place) ----------------
__global__ void pres_kernel(float* __restrict__ dot, const float* __restrict__ nullv,
                            const unsigned char* __restrict__ mask, int nPer, long total)
{
    long e = (long)blockIdx.x * blockDim.x + threadIdx.x;
    if (e >= total) return;
    int k = (int)(e % nPer);
    long row = e / nPer;
    int b = (int)(row / Nn);
    float v = 0.0f;
    if (mask[b * nPer + k]) v = 1.0f / (1.0f + expf(-(dot[e] - nullv[row])));
    dot[e] = v;
}

// ---------------- one-hot counts + class-mask count ----------------
__global__ void counts_kernel(const int* __restrict__ idx, const unsigned char* __restrict__ imask,
                              const unsigned char* __restrict__ cmask,
                              float* __restrict__ counts, float* __restrict__ accB)
{
    int b = blockIdx.x, t = threadIdx.x; // 128 threads
    if (t < Kk) {
        float c = 0.0f;
        for (int i = 0; i < Ii; ++i) {
            if (imask[b * Ii + i]) {
                int k = idx[b * Ii + i];
                k = k < 0 ? 0 : (k > Kk - 1 ? Kk - 1 : k);
                if (k == t) c += 1.0f;
            }
        }
        counts[b * Kk + t] = fmaxf(c, 1.0f);
    }
    if (t == 0) {
        float s = 0.0f;
        for (int k = 0; k < Kk; ++k) s += cmask[b * Kk + k] ? 1.0f : 0.0f;
        accB[b] = fmaxf(s, 1.0f);
    }
}

// ---------------- instance aggregation + combined + row reductions ----------------
__global__ void agg_kernel(const float* __restrict__ instPres, const float* __restrict__ classPres,
                           const float* __restrict__ counts, const unsigned char* __restrict__ cmask,
                           const int* __restrict__ idx, const unsigned char* __restrict__ imask,
                           float* __restrict__ combined, float* __restrict__ combMax,
                           float* __restrict__ combSum, float* __restrict__ instMax)
{
    __shared__ float acc[Kk];
    __shared__ float red[128];
    int row = blockIdx.x, b = row / Nn, t = threadIdx.x; // 128 threads
    if (t < Kk) acc[t] = 0.0f;
    __syncthreads();
    float pm = -1e30f;
    for (int i = t; i < Ii; i += 128) {
        float v = instPres[(long)row * Ii + i];
        pm = fmaxf(pm, v);
        if (imask[b * Ii + i]) {
            int k = idx[b * Ii + i];
            k = k < 0 ? 0 : (k > Kk - 1 ? Kk - 1 : k);
            atomicAdd(&acc[k], v);
        }
    }
    red[t] = pm; __syncthreads();
    for (int s = 64; s > 0; s >>= 1) { if (t < s) red[t] = fmaxf(red[t], red[t + s]); __syncthreads(); }
    if (t < Kk) {
        float cip = cmask[b * Kk + t] ? (acc[t] / counts[b * Kk + t]) : 0.0f;
        float cb = 0.5f * (classPres[(long)row * Kk + t] + cip);
        combined[(long)row * Kk + t] = cb;
        acc[t] = cb;
    }
    __syncthreads();
    if (t == 0) {
        float cm = -1e30f, cs = 0.0f;
        for (int k = 0; k < Kk; ++k) { cm = fmaxf(cm, acc[k]); cs += acc[k]; }
        combMax[row] = cm; combSum[row] = cs; instMax[row] = red[0];
    }
}

// ---------------- FiLM-style branch fuse ----------------
__global__ void branch_kernel(const float* __restrict__ q2, const float* __restrict__ gb,
                              const float* __restrict__ combined, const float* __restrict__ accB,
                              float* __restrict__ q3)
{
    __shared__ float sc[Kk];
    int row = blockIdx.x, b = row / Nn, c = threadIdx.x; // 256 threads
    if (c < Kk) sc[c] = combined[(long)row * Kk + c];
    __syncthreads();
    float qv = q2[(long)row * Cc + c];
    float mx = -1e30f, sm = 0.0f;
    for (int k = 0; k < Kk; ++k) {
        const float* g = gb + ((long)(b * Kk + k)) * (2 * Cc);
        float gamma = tanhf(g[c]);
        float beta  = g[Cc + c];
        float v = (qv * (1.0f + gamma) + beta) * sc[k];
        mx = fmaxf(mx, v);
        sm += v;
    }
    q3[(long)row * Cc + c] = qv + 0.5f * (mx + sm / accB[b]);
}

// ---------------- assemble 262-channel maps with 1px zero halo (B,262,50,50) ----------------
__global__ void maps_kernel(const float* __restrict__ q3, const float* __restrict__ combMax,
                            const float* __restrict__ combSum, const float* __restrict__ sCombMax,
                            const float* __restrict__ sCombSum, const float* __restrict__ instMax,
                            const float* __restrict__ instSimMax, const float* __restrict__ accB,
                            float* __restrict__ maps)
{
    long e = (long)blockIdx.x * blockDim.x + threadIdx.x;
    long total = (long)Bb_ * MAPC * PA;
    if (e >= total) return;
    int xx = (int)(e % PH);
    long r2 = e / PH;
    int yy = (int)(r2 % PH);
    long r3 = r2 / PH;
    int ch = (int)(r3 % MAPC);
    int b  = (int)(r3 / MAPC);
    float v = 0.0f;
    if (xx > 0 && xx < PH - 1 && yy > 0 && yy < PH - 1) {
        int hw = (yy - 1) * Ww + (xx - 1);
        long row = (long)b * Nn + hw;
        if (ch < Cc)           v = q3[row * Cc + ch];
        else if (ch == Cc)     v = combMax[row];
        else if (ch == Cc + 1) v = combSum[row] / accB[b];
        else if (ch == Cc + 2) v = sCombMax[row];
        else if (ch == Cc + 3) v = sCombSum[row] / accB[b];
        else if (ch == Cc + 4) v = instMax[row];
        else                   v = instSimMax[row];
    }
    maps[e] = v;
}

// ---------------- conv weight repack: Wr(col,ic,ky,kx) -> Wp[col][r][icp] ----------------
// icp padded MAPC->ICP with zeros so the conv K loop needs no weight guards.
__global__ void wpack_kernel(const float* __restrict__ Wr, float* __restrict__ Wp)
{
    long e = (long)blockIdx.x * blockDim.x + threadIdx.x;
    long total = (long)Cc * 9 * ICP;
    if (e >= total) return;
    int icp = (int)(e % ICP);
    long r2 = e / ICP;
    int r   = (int)(r2 % 9);
    int col = (int)(r2 / 9);
    Wp[e] = (icp < MAPC) ? Wr[((long)col * MAPC + icp) * 9 + r] : 0.0f;
}

// ---------------- 3x3 conv as implicit-im2col WMMA GEMM + BN + ReLU ----------------
// K ordered (tap r, channel ic): for fixed tap, A address = base + ic*PA
// (pure strided), W loads contiguous b64 from the packed layout. The ic pad
// (262..271) multiplies zero weights, so A just clamps the index.
__global__ void conv_wmma(const float* __restrict__ mapsPad, const float* __restrict__ Wp,
                          const float* __restrict__ br, const float* __restrict__ bng,
                          const float* __restrict__ bnb, float* __restrict__ out)
{
    int lane = threadIdx.x;
    int hl = lane >> 4, l16 = lane & 15;
    int tN = blockIdx.x, tM = blockIdx.y;
    int mrow = tM * 16 + l16;                 // pixel id in [0, B*n)
    int b = mrow / Nn, hw = mrow % Nn;
    int y = hw / Ww, x = hw % Ww;
    int col = tN * 16 + l16;                  // out channel (<256, exact)
    const float* Abase = mapsPad + (long)b * MAPC * PA + (long)y * PH + x; // halo absorbs -1
    const float* Wcol  = Wp + (long)col * (9 * ICP);
    v8f acc = {};
    for (int r = 0; r < 9; ++r) {             // 3x3 tap decoded ONCE per tap
        int ky = r / 3, kx = r - ky * 3;
        const float* Ar = Abase + ky * PH + kx;
        const float* Wrp = Wcol + r * ICP;
        for (int ic0 = 0; ic0 < ICP; ic0 += 16) {
            v2f a[4], w[4];
#pragma unroll
            for (int u = 0; u < 4; ++u) {
                int ica = ic0 + u * 4 + hl * 2;
                int i0 = ica     < MAPC ? ica     : MAPC - 1;  // pad lanes hit zero weight
                int i1 = ica + 1 < MAPC ? ica + 1 : MAPC - 1;
                a[u].x = Ar[(long)i0 * PA];
                a[u].y = Ar[(long)i1 * PA];
                w[u] = *(const v2f*)(Wrp + ica);
            }
#pragma unroll
            for (int u = 0; u < 4; ++u)
                acc = __builtin_amdgcn_wmma_f32_16x16x4_f32(false, a[u], false, w[u], (short)0,
                                                            acc, false, false);
        }
    }
#pragma unroll
    for (int j = 0; j < 8; ++j) {
        int r = tM * 16 + hl * 8 + j;
        int bb = r / Nn, hh = r % Nn;
        float v = acc[j] + br[col];
        v = v * bng[col] + bnb[col];
        v = fmaxf(v, 0.0f);
        out[((long)bb * Cc + col) * Nn + hh] = v;
    }
}

// ===================== host-side orchestration =====================
static inline float* wsalloc(char*& p, size_t nfloats)
{
    float* r = (float*)p;
    p += ((nfloats * sizeof(float) + 255) & ~(size_t)255);
    return r;
}

extern "C" void kernel_launch(void* const* d_in, const int* in_sizes, int n_in,
                              void* d_out, int out_size, void* d_ws, size_t ws_size,
                              hipStream_t stream)
{
    const float*         x       = (const float*)d_in[0];
    const float*         prompt  = (const float*)d_in[1];
    const unsigned char* pmask   = (const unsigned char*)d_in[2];
    const float*         cproto  = (const float*)d_in[3];
    const unsigned char* cmask   = (const unsigned char*)d_in[4];
    const float*         stok    = (const float*)d_in[5];
    const float*         iproto  = (const float*)d_in[6];
    const int*           iidx    = (const int*)d_in[7];
    const unsigned char* imask   = (const unsigned char*)d_in[8];
    const float*         istok   = (const float*)d_in[9];
    const float* qn_g = (const float*)d_in[10], *qn_b = (const float*)d_in[11];
    const float* pn_g = (const float*)d_in[12], *pn_b = (const float*)d_in[13];
    const float* Wq = (const float*)d_in[14], *bq = (const float*)d_in[15];
    const float* Wk = (const float*)d_in[16], *bk = (const float*)d_in[17];
    const float* Wv = (const float*)d_in[18], *bv = (const float*)d_in[19];
    const float* Wo = (const float*)d_in[20], *bo = (const float*)d_in[21];
    const float* W1 = (const float*)d_in[22], *b1 = (const float*)d_in[23];
    const float* W2 = (const float*)d_in[24], *b2 = (const float*)d_in[25];
    const float* Wqp = (const float*)d_in[26], *bqp = (const float*)d_in[27];
    const float* Wf = (const float*)d_in[28], *bf = (const float*)d_in[29];
    const float* Wsim = (const float*)d_in[30], *bsim = (const float*)d_in[31];
    const float* nullp = (const float*)d_in[32], *nulls = (const float*)d_in[33];
    const float* Wr = (const float*)d_in[34], *br = (const float*)d_in[35];
    const float* bn_g = (const float*)d_in[36], *bn_b = (const float*)d_in[37];

    const int BN = Bb_ * Nn;   // 4608
    const int BL = Bb_ * Ll;   // 154

    char* p = (char*)d_ws;
    float* q0      = wsalloc(p, (size_t)BN * Cc);
    float* ptb     = wsalloc(p, (size_t)BL * Cc);
    float* Qb      = wsalloc(p, (size_t)BN * Cc);
    float* Kb      = wsalloc(p, (size_t)BL * Cc);
    float* Vb      = wsalloc(p, (size_t)BL * Cc);
    float* ao      = wsalloc(p, (size_t)BN * Cc);
    float* q1      = wsalloc(p, (size_t)BN * Cc);
    float* hb      = wsalloc(p, (size_t)BN * C4);
    float* q2      = wsalloc(p, (size_t)BN * Cc);
    float* qp      = wsalloc(p, (size_t)BN * Pp);
    float* protoN  = wsalloc(p, (size_t)Bb_ * Kk * Pp);
    float* instN   = wsalloc(p, (size_t)Bb_ * Ii * Pp);
    float* nullpN  = wsalloc(p, Pp);
    float* null_l  = wsalloc(p, BN);
    float* clsd    = wsalloc(p, (size_t)BN * Kk);   // -> class_pres in place
    float* instd   = wsalloc(p, (size_t)BN * Ii);   // -> inst_pres in place
    float* counts  = wsalloc(p, Bb_ * Kk);
    float* accB    = wsalloc(p, Bb_);
    float* combined= wsalloc(p, (size_t)BN * Kk);
    float* combMax = wsalloc(p, BN);
    float* combSum = wsalloc(p, BN);
    float* instMaxB= wsalloc(p, BN);
    float* gbuf    = wsalloc(p, (size_t)Bb_ * Kk * 2 * Cc);
    float* q3      = wsalloc(p, (size_t)BN * Cc);
    float* ft      = wsalloc(p, (size_t)BN * Cc);
    float* scaleN  = wsalloc(p, (size_t)Bb_ * Kk * Cc);
    float* iscaleN = wsalloc(p, (size_t)Bb_ * Ii * Cc);
    float* nullsN  = wsalloc(p, Cc);
    float* nsl     = wsalloc(p, BN);
    float* sclsd   = wsalloc(p, (size_t)BN * Kk);
    float* sinstd  = wsalloc(p, (size_t)BN * Ii);
    float* combinedS = wsalloc(p, (size_t)BN * Kk);
    float* sCombMax  = wsalloc(p, BN);
    float* sCombSum  = wsalloc(p, BN);
    float* instSimMaxB = wsalloc(p, BN);
    float* mapsb   = wsalloc(p, (size_t)Bb_ * MAPC * PA);  // padded (B,262,50,50)
    // transposed weights (N, K) and packed conv weights
    float* WqT   = wsalloc(p, (size_t)Cc * Cc);
    float* WkT   = wsalloc(p, (size_t)Cc * Cc);
    float* WvT   = wsalloc(p, (size_t)Cc * Cc);
    float* WoT   = wsalloc(p, (size_t)Cc * Cc);
    float* W1T   = wsalloc(p, (size_t)Cc * C4);
    float* W2T   = wsalloc(p, (size_t)C4 * Cc);
    float* WqpT  = wsalloc(p, (size_t)Cc * Pp);
    float* WfT   = wsalloc(p, (size_t)Cc * 2 * Cc);
    float* WsimT = wsalloc(p, (size_t)Cc * Cc);
    float* Wpack = wsalloc(p, (size_t)Cc * 9 * ICP);

    auto transpose = [&](const float* in, float* out, int K, int N) {
        long tot = (long)K * N;
        transpose_kernel<<<(int)((tot + 255) / 256), 256, 0, stream>>>(in, out, K, N);
    };
    auto gemm = [&](const float* A, const float* Bt, const float* bias, const float* resid,
                    float* D, int M, int N, int Kd, int act,
                    long sA, long sB, long sD, int batch) {
        dim3 g((N + 15) / 16, (M + 15) / 16, batch);
        gemm_wmma<<<g, 32, 0, stream>>>(A, Bt, bias, resid, D, M, N, Kd, act, sA, sB, sD);
    };

    // 0) weight repacks (K-major B operands for all GEMMs; padded conv pack)
    transpose(Wq,  WqT,  Cc, Cc);
    transpose(Wk,  WkT,  Cc, Cc);
    transpose(Wv,  WvT,  Cc, Cc);
    transpose(Wo,  WoT,  Cc, Cc);
    transpose(W1,  W1T,  Cc, C4);
    transpose(W2,  W2T,  C4, Cc);
    transpose(Wqp, WqpT, Cc, Pp);
    transpose(Wf,  WfT,  Cc, 2 * Cc);
    transpose(Wsim, WsimT, Cc, Cc);
    {
        long tot = (long)Cc * 9 * ICP;
        wpack_kernel<<<(int)((tot + 255) / 256), 256, 0, stream>>>(Wr, Wpack);
    }

    // 1) LayerNorms
    ln_kernel<<<BN, 256, 0, stream>>>(x, qn_g, qn_b, q0, 1);
    ln_kernel<<<BL, 256, 0, stream>>>(prompt, pn_g, pn_b, ptb, 0);
    // 2) Q/K/V projections
    gemm(q0,  WqT, bq, nullptr, Qb, BN, Cc, Cc, 0, 0, 0, 0, 1);
    gemm(ptb, WkT, bk, nullptr, Kb, BL, Cc, Cc, 0, 0, 0, 0, 1);
    gemm(ptb, WvT, bv, nullptr, Vb, BL, Cc, Cc, 0, 0, 0, 0, 1);
    // 3) attention + output proj (+residual)
    attn_kernel<<<BN, 256, 0, stream>>>(Qb, Kb, Vb, pmask, ao);
    gemm(ao, WoT, bo, q0, q1, BN, Cc, Cc, 0, 0, 0, 0, 1);
    // 4) MLP (+residual)
    gemm(q1, W1T, b1, nullptr, hb, BN, C4, Cc, 1, 0, 0, 0, 1);
    gemm(hb, W2T, b2, q1, q2, BN, Cc, C4, 0, 0, 0, 0, 1);
    // 5) query prototypes + normalizations
    gemm(q2, WqpT, bqp, nullptr, qp, BN, Pp, Cc, 0, 0, 0, 0, 1);
    norm_kernel<<<BN, 256, 0, stream>>>(qp, qp, Pp);
    norm_kernel<<<Bb_ * Kk, 256, 0, stream>>>(cproto, protoN, Pp);
    norm_kernel<<<Bb_ * Ii, 256, 0, stream>>>(iproto, instN, Pp);
    norm_kernel<<<1, 256, 0, stream>>>(nullp, nullpN, Pp);
    rowdot_kernel<<<BN, 256, 0, stream>>>(qp, nullpN, null_l, Pp);
    // 6) class / instance similarity (batched WMMA GEMM; prototypes already (N,K))
    gemm(qp, protoN, nullptr, nullptr, clsd, Nn, Kk, Pp, 0,
         (long)Nn * Pp, (long)Kk * Pp, (long)Nn * Kk, Bb_);
    gemm(qp, instN, nullptr, nullptr, instd, Nn, Ii, Pp, 0,
         (long)Nn * Pp, (long)Ii * Pp, (long)Nn * Ii, Bb_);
    {
        long tc = (long)BN * Kk, ti = (long)BN * Ii;
        pres_kernel<<<(int)((tc + 255) / 256), 256, 0, stream>>>(clsd, null_l, cmask, Kk, tc);
        pres_kernel<<<(int)((ti + 255) / 256), 256, 0, stream>>>(instd, null_l, imask, Ii, ti);
    }
    counts_kernel<<<Bb_, 128, 0, stream>>>(iidx, imask, cmask, counts, accB);
    agg_kernel<<<BN, 128, 0, stream>>>(instd, clsd, counts, cmask, iidx, imask,
                                       combined, combMax, combSum, instMaxB);
    // 7) FiLM params + branch fuse
    gemm(cproto, WfT, bf, nullptr, gbuf, Bb_ * Kk, 2 * Cc, Pp, 0, 0, 0, 0, 1);
    branch_kernel<<<BN, 256, 0, stream>>>(q2, gbuf, combined, accB, q3);
    // 8) sim features
    gemm(q3, WsimT, bsim, nullptr, ft, BN, Cc, Cc, 0, 0, 0, 0, 1);
    norm_kernel<<<BN, 256, 0, stream>>>(ft, ft, Cc);
    norm_kernel<<<Bb_ * Kk, 256, 0, stream>>>(stok, scaleN, Cc);
    norm_kernel<<<Bb_ * Ii, 256, 0, stream>>>(istok, iscaleN, Cc);
    norm_kernel<<<1, 256, 0, stream>>>(nulls, nullsN, Cc);
    rowdot_kernel<<<BN, 256, 0, stream>>>(ft, nullsN, nsl, Cc);
    gemm(ft, scaleN, nullptr, nullptr, sclsd, Nn, Kk, Cc, 0,
         (long)Nn * Cc, (long)Kk * Cc, (long)Nn * Kk, Bb_);
    gemm(ft, iscaleN, nullptr, nullptr, sinstd, Nn, Ii, Cc, 0,
         (long)Nn * Cc, (long)Ii * Cc, (long)Nn * Ii, Bb_);
    {
        long tc = (long)BN * Kk, ti = (long)BN * Ii;
        pres_kernel<<<(int)((tc + 255) / 256), 256, 0, stream>>>(sclsd, nsl, cmask, Kk, tc);
        pres_kernel<<<(int)((ti + 255) / 256), 256, 0, stream>>>(sinstd, nsl, imask, Ii, ti);
    }
    agg_kernel<<<BN, 128, 0, stream>>>(sinstd, sclsd, counts, cmask, iidx, imask,
                                       combinedS, sCombMax, sCombSum, instSimMaxB);
    // 9) padded maps + conv head
    {
        long total = (long)Bb_ * MAPC * PA;
        maps_kernel<<<(int)((total + 255) / 256), 256, 0, stream>>>(
            q3, combMax, combSum, sCombMax, sCombSum, instMaxB, instSimMaxB, accB, mapsb);
    }
    conv_wmma<<<dim3(Cc / 16, BN / 16), 32, 0, stream>>>(mapsb, Wpack, br, bn_g, bn_b,
                                                         (float*)d_out);

    (void)in_sizes; (void)n_in; (void)out_size; (void)ws_size;
}